// NLL_87333864997268
// MI455X (gfx1250) — compile-verified
//
#include <hip/hip_runtime.h>
#include <cstdint>

typedef __attribute__((ext_vector_type(2))) float v2f;
typedef __attribute__((ext_vector_type(8))) float v8f;

#define QSEG 100000
#define ACC_STRIDE 8   // yf, sy, lg0..lg4, pad -> one 32B cacheline chunk per segment

// Gauss-Hermite (n=5) nodes and weights*1/sqrt(pi)
#define XK0 2.0201828704560856f
#define XK1 0.9585724646138185f
#define WH2 0.011257411327720693f   // wk(|x|=2.0202)/sqrt(pi)
#define WH1 0.22207592200561263f    // wk(|x|=0.9586)/sqrt(pi)
#define WH0 0.53333333333333333f    // wk(0)/sqrt(pi) = 8/15

__device__ __forceinline__ float softplus_f(float x) {
    // logaddexp(x, 0) = max(x,0) + log1p(exp(-|x|)); exp(-|x|) in (0,1] so log(1+e) is safe
    return fmaxf(x, 0.0f) + __logf(1.0f + __expf(-fabsf(x)));
}

__device__ __forceinline__ void atomic_add_f32(float* p, float v) {
    // relaxed + agent scope -> hardware global_atomic_add_f32 (accumulator lives in L2)
    __hip_atomic_fetch_add(p, v, __ATOMIC_RELAXED, __HIP_MEMORY_SCOPE_AGENT);
}

__global__ void nll_zero_kernel(float* __restrict__ acc, int n4, float* __restrict__ out) {
    int i = blockIdx.x * blockDim.x + threadIdx.x;
    if (i < n4) ((float4*)acc)[i] = make_float4(0.0f, 0.0f, 0.0f, 0.0f);
    if (i == 0) out[0] = 0.0f;
}

__global__ void nll_scatter_kernel(const float* __restrict__ y,
                                   const float* __restrict__ f,
                                   const int* __restrict__ seg,
                                   const float* __restrict__ sig2b,
                                   float* __restrict__ acc, int n) {
    __shared__ float ly[256];
    __shared__ float lf[256];
    const int tid = threadIdx.x;
    const int i = blockIdx.x * 256 + tid;
    if (i >= n) return;

    // gfx1250 async global->LDS staging (ASYNCcnt path): each lane stages its own
    // y/f element into its private LDS slot while we fetch seg and compute ck.
    {
        unsigned ldsy = (unsigned)(uintptr_t)&ly[tid];
        unsigned ldsf = (unsigned)(uintptr_t)&lf[tid];
        unsigned long long ay = (unsigned long long)(uintptr_t)(y + i);
        unsigned long long af = (unsigned long long)(uintptr_t)(f + i);
        asm volatile("global_load_async_to_lds_b32 %0, %1, off"
                     :: "v"(ldsy), "v"(ay) : "memory");
        asm volatile("global_load_async_to_lds_b32 %0, %1, off"
                     :: "v"(ldsf), "v"(af) : "memory");
    }

    const int q = seg[i];
    const float c = sqrtf(2.0f * sig2b[0]);     // ck_k = c * xk_k

    asm volatile("s_wait_asynccnt 0" ::: "memory");
    const float yv = ly[tid];
    const float fv = lf[tid];

    float* base = acc + (size_t)q * ACC_STRIDE;
    atomic_add_f32(base + 0, yv * fv);                      // yf
    atomic_add_f32(base + 1, yv);                           // sy
    atomic_add_f32(base + 2, softplus_f(fv - c * XK0));     // lg[k], nodes ascending
    atomic_add_f32(base + 3, softplus_f(fv - c * XK1));
    atomic_add_f32(base + 4, softplus_f(fv));
    atomic_add_f32(base + 5, softplus_f(fv + c * XK1));
    atomic_add_f32(base + 6, softplus_f(fv + c * XK0));
}

__global__ void nll_combine_kernel(const float* __restrict__ acc,
                                   const float* __restrict__ sig2b,
                                   float* __restrict__ out) {
    const int tid = threadIdx.x;
    const int q = blockIdx.x * 256 + tid;

    float t = 0.0f;  // out-of-range lanes contribute 0 but stay active (WMMA needs full EXEC)
    if (q < QSEG) {
        const float4 v0 = ((const float4*)acc)[q * 2];      // yf, sy, lg0, lg1
        const float4 v1 = ((const float4*)acc)[q * 2 + 1];  // lg2, lg3, lg4, pad
        const float c  = sqrtf(2.0f * sig2b[0]);
        const float yf = v0.x, syc0 = v0.y * c * XK0, syc1 = v0.y * c * XK1;
        float ks = WH2 * __expf(yf - syc0 - v0.z)
                 + WH1 * __expf(yf - syc1 - v0.w)
                 + WH0 * __expf(yf        - v1.x)
                 + WH1 * __expf(yf + syc1 - v1.y)
                 + WH2 * __expf(yf + syc0 - v1.z);
        t = -__logf(ks);
    }

    // Wave-level sum on the matrix pipe: A(16x4) holds lane values in K-columns 0/2
    // (lanes 0-15 -> K=0, lanes 16-31 -> K=2), B = ones(4x16).
    // D[m,n] = s[m] + s[m+16] for lanes 0-15, s[m+8] + s[m+24] for lanes 16-31.
    v2f a; a.x = t;    a.y = 0.0f;
    v2f b; b.x = 1.0f; b.y = 1.0f;
    v8f cacc = {};
    v8f d = __builtin_amdgcn_wmma_f32_16x16x4_f32(false, a, false, b, (short)0, cacc,
                                                  false, false);
    float u = d[0] + d[1] + d[2] + d[3] + d[4] + d[5] + d[6] + d[7];
    u += __shfl_xor(u, 16, 32);          // combine low/high half-wave partials
    if ((tid & 31) == 0) atomic_add_f32(out, u);
}

extern "C" void kernel_launch(void* const* d_in, const int* in_sizes, int n_in,
                              void* d_out, int out_size, void* d_ws, size_t ws_size,
                              hipStream_t stream) {
    (void)n_in; (void)out_size; (void)ws_size;
    const float* y_true = (const float*)d_in[0];
    const float* y_pred = (const float*)d_in[1];
    const int*   Z_idx  = (const int*)d_in[2];
    const float* sig2b  = (const float*)d_in[3];
    float* out = (float*)d_out;
    float* acc = (float*)d_ws;   // QSEG * ACC_STRIDE floats = 3.2 MB (fits in L2)
    const int n  = in_sizes[0];
    const int n4 = (QSEG * ACC_STRIDE) / 4;

    nll_zero_kernel   <<<(n4 + 255) / 256, 256, 0, stream>>>(acc, n4, out);
    nll_scatter_kernel<<<(n  + 255) / 256, 256, 0, stream>>>(y_true, y_pred, Z_idx,
                                                             sig2b, acc, n);
    nll_combine_kernel<<<(QSEG + 255) / 256, 256, 0, stream>>>(acc, sig2b, out);
}